// LatticeVariance_86517821214970
// MI455X (gfx1250) — compile-verified
//
#include <hip/hip_runtime.h>
#include <hip/hip_bf16.h>

// Problem constants (from reference): B=2, GH=GW=17, H=W=192, C=64, K=512
#define BB    2
#define HH    192
#define WW    192
#define HWPIX (HH*WW)        // 36864
#define CC    64
#define KTRI  512
#define LDSS  520            // 512 + 8 pad -> avoid LDS bank conflicts between rows

typedef __attribute__((ext_vector_type(2))) float v2f;
typedef __attribute__((ext_vector_type(8))) float v8f;

__device__ __forceinline__ int imin(int a, int b) { return a < b ? a : b; }

// ---------------------------------------------------------------------------
// Kernel 0: zero segment sums + counts
// ---------------------------------------------------------------------------
__global__ void lv_zero(float* __restrict__ sums, int* __restrict__ cnt) {
    int i = blockIdx.x * blockDim.x + threadIdx.x;
    if (i < BB * KTRI * CC) sums[i] = 0.0f;
    if (i < BB * KTRI)      cnt[i]  = 0;
}

// ---------------------------------------------------------------------------
// Kernel 1: per-triangle edge coefficients.
// d_e(p) = A_e*px + B_e*py + C_e ; coef layout [b][edge*3+row][KTRI]
// ---------------------------------------------------------------------------
__global__ void lv_coef(const float* __restrict__ grid_pos, // [B][289][2]
                        const int*   __restrict__ tri,      // [B][K][3]
                        float*       __restrict__ coef) {   // [B][9][K]
    int idx = blockIdx.x * blockDim.x + threadIdx.x;
    if (idx >= BB * KTRI) return;
    int b = idx >> 9, t = idx & (KTRI - 1);
    const int*   tv = tri + (b * KTRI + t) * 3;
    const float* gp = grid_pos + b * 289 * 2;
    float ax = gp[tv[0]*2], ay = gp[tv[0]*2+1];
    float bx = gp[tv[1]*2], by = gp[tv[1]*2+1];
    float cx = gp[tv[2]*2], cy = gp[tv[2]*2+1];
    // ed(p,v1,v2) = (px-v2x)(v1y-v2y) - (v1x-v2x)(py-v2y)
    float A1 = ay - by, B1 = -(ax - bx), C1 = -(bx * A1 + by * B1); // (a,b)
    float A2 = by - cy, B2 = -(bx - cx), C2 = -(cx * A2 + cy * B2); // (b,c)
    float A3 = cy - ay, B3 = -(cx - ax), C3 = -(ax * A3 + ay * B3); // (c,a)
    float* cf = coef + b * 9 * KTRI;
    cf[0*KTRI+t] = A1; cf[1*KTRI+t] = B1; cf[2*KTRI+t] = C1;
    cf[3*KTRI+t] = A2; cf[4*KTRI+t] = B2; cf[5*KTRI+t] = C2;
    cf[6*KTRI+t] = A3; cf[7*KTRI+t] = B3; cf[8*KTRI+t] = C3;
}

// ---------------------------------------------------------------------------
// Kernel 2: cond via V_WMMA_F32_16X16X4_F32.
// A (16x4) = [px, py, 1, 0] for a 16-pixel tile; B (4x16) = one edge's
// [A;B;C;0] coefficients for 16 triangles. Three WMMAs per chunk put
// d1,d2,d3 of one (pixel,triangle) pair in the same lane of 3 accumulators.
// All LDS loads are unconditional (value-selected with cndmask) so EXEC
// stays all-ones through the entire WMMA loop — no divergent regions.
// 8 waves / block, one 16-pixel tile per wave, 288 blocks per batch.
// ---------------------------------------------------------------------------
__global__ __launch_bounds__(256) void lv_cond(const float* __restrict__ coef,
                                               int*   __restrict__ cond_i,
                                               float* __restrict__ cond_f) {
    __shared__ float lds[9 * LDSS];
    int b         = blockIdx.x / 288;
    int blockTile = blockIdx.x % 288;

    const float* cb = coef + b * 9 * KTRI;
    for (int i = threadIdx.x; i < 9 * KTRI; i += 256) {
        int row = i >> 9, t = i & (KTRI - 1);
        lds[row * LDSS + t] = cb[i];
    }
    __syncthreads();

    int  lane = threadIdx.x & 31;
    int  wave = threadIdx.x >> 5;
    int  tileBase = (blockTile * 8 + wave) * 16;   // pixel base within batch
    int  m  = lane & 15;
    bool lo = lane < 16;

    // A-matrix (ISA 16x4 f32 layout): lanes 0-15 VGPR0=K0(px) VGPR1=K1(py);
    // lanes 16-31 VGPR0=K2(1.0) VGPR1=K3(0.0), rows M = lane&15.
    int   p  = tileBase + m;
    float px = ((float)(p % WW) + 0.5f) * (1.0f / WW);
    float py = ((float)(p / WW) + 0.5f) * (1.0f / HH);
    v2f a;
    a.x = lo ? px : 1.0f;
    a.y = lo ? py : 0.0f;

    // Loop-invariant per-lane LDS base offsets (row select is arithmetic).
    int rsel = lo ? 0 : 2;
    int base0_0 = (0 + rsel) * LDSS + m;   // edge 0: row A (lo) / row C (hi)
    int base0_1 = (3 + rsel) * LDSS + m;   // edge 1
    int base0_2 = (6 + rsel) * LDSS + m;   // edge 2
    int base1_0 = 1 * LDSS + m;            // edge 0: row B (zeroed on hi lanes)
    int base1_1 = 4 * LDSS + m;
    int base1_2 = 7 * LDSS + m;

    int best[8];
#pragma unroll
    for (int r = 0; r < 8; ++r) best[r] = 0x7fffffff;

#pragma unroll 2
    for (int tbase = 0; tbase < KTRI; tbase += 16) {
        v8f d0, d1, d2;
#pragma unroll
        for (int e = 0; e < 3; ++e) {
            int b0i = (e == 0 ? base0_0 : (e == 1 ? base0_1 : base0_2)) + tbase;
            int b1i = (e == 0 ? base1_0 : (e == 1 ? base1_1 : base1_2)) + tbase;
            float b0  = lds[b0i];
            float b1v = lds[b1i];                 // unconditional load
            float b1  = lo ? b1v : 0.0f;          // cndmask, no branch
            v2f bm; bm.x = b0; bm.y = b1;
            v8f cz = {};
            v8f acc = __builtin_amdgcn_wmma_f32_16x16x4_f32(
                false, a, false, bm, (short)0, cz, false, false);
            if (e == 0) d0 = acc; else if (e == 1) d1 = acc; else d2 = acc;
        }
#pragma unroll
        for (int r = 0; r < 8; ++r) {
            float e1 = d0[r], e2 = d1[r], e3 = d2[r];
            bool neg = (e1 < 0.0f) | (e2 < 0.0f) | (e3 < 0.0f);
            bool pos = (e1 > 0.0f) | (e2 > 0.0f) | (e3 > 0.0f);
            bool inside = !(neg & pos);
            int cand = inside ? (tbase + m) : 0x7fffffff;
            best[r] = imin(best[r], cand);
        }
    }

    // min across the 16 triangle-columns (xor masks <16 stay within each half)
#pragma unroll
    for (int r = 0; r < 8; ++r) {
        int v = best[r];
        v = imin(v, __shfl_xor(v, 1, 32));
        v = imin(v, __shfl_xor(v, 2, 32));
        v = imin(v, __shfl_xor(v, 4, 32));
        v = imin(v, __shfl_xor(v, 8, 32));
        best[r] = v;
    }
    // D layout: lanes 0-15 row r = pixel tileBase+r ; lanes 16-31 = tileBase+8+r
    if (m == 0) {
        int off = lo ? 0 : 8;
#pragma unroll
        for (int r = 0; r < 8; ++r) {
            int c  = best[r];
            int cv = (c == 0x7fffffff) ? -1 : c;
            int pix = b * HWPIX + tileBase + off + r;
            cond_i[pix] = cv;
            cond_f[pix] = (float)cv;
        }
    }
}

// ---------------------------------------------------------------------------
// Kernel 3: segment sums (float atomics; lanes hit distinct channels) + counts
// ---------------------------------------------------------------------------
__global__ void lv_accum(const float* __restrict__ fea, const int* __restrict__ cond_i,
                         float* __restrict__ sums, int* __restrict__ cnt) {
    int idx = blockIdx.x * blockDim.x + threadIdx.x;   // (b,p,c), c fastest
    if (idx >= BB * HWPIX * CC) return;
    int c  = idx & (CC - 1);
    int bp = idx >> 6;
    int s  = cond_i[bp];
    if (s < 0) return;
    int b = bp / HWPIX;
    atomicAdd(&sums[(b * KTRI + s) * CC + c], fea[idx]);
    if (c == 0) atomicAdd(&cnt[b * KTRI + s], 1);
}

// ---------------------------------------------------------------------------
// Kernel 4: grid_fea = sums / max(cnt,1)  -> d_out region
// ---------------------------------------------------------------------------
__global__ void lv_gridfea(const float* __restrict__ sums, const int* __restrict__ cnt,
                           float* __restrict__ out_gf) {
    int idx = blockIdx.x * blockDim.x + threadIdx.x;
    if (idx >= BB * KTRI * CC) return;
    float n = fmaxf((float)cnt[idx >> 6], 1.0f);
    out_gf[idx] = sums[idx] / n;
}

// ---------------------------------------------------------------------------
// Kernel 5: recon gather + per-block partial (sum_sq, sum_abs).
// One wave per pixel, 2 channels per lane (coalesced 256B/wave).
// 8 pixels per block; 4608 blocks per batch (batch-aligned).
// ---------------------------------------------------------------------------
__global__ __launch_bounds__(256) void lv_recon(const float* __restrict__ fea,
                                                const int*   __restrict__ cond_i,
                                                const float* __restrict__ gridfea,
                                                float* __restrict__ recon,
                                                float* __restrict__ partials) {
    __shared__ float ssq[8], sab[8];
    int lane = threadIdx.x & 31;
    int wave = threadIdx.x >> 5;
    int bp   = blockIdx.x * 8 + wave;          // (b*HWPIX + p)
    int b    = bp / HWPIX;
    int s    = cond_i[bp];

    int base = bp * CC + lane * 2;
    float f0 = fea[base], f1 = fea[base + 1];
    float g0 = 0.0f, g1 = 0.0f;
    if (s >= 0) {
        int gbase = (b * KTRI + s) * CC + lane * 2;
        g0 = gridfea[gbase];
        g1 = gridfea[gbase + 1];
    }
    recon[base]     = g0;
    recon[base + 1] = g1;

    float e0 = f0 - g0, e1 = f1 - g1;
    float sq = e0 * e0 + e1 * e1;
    float ab = fabsf(e0) + fabsf(e1);
    for (int off = 16; off; off >>= 1) {
        sq += __shfl_xor(sq, off, 32);
        ab += __shfl_xor(ab, off, 32);
    }
    if (lane == 0) { ssq[wave] = sq; sab[wave] = ab; }
    __syncthreads();
    if (threadIdx.x == 0) {
        float S = 0.0f, A = 0.0f;
        for (int r = 0; r < 8; ++r) { S += ssq[r]; A += sab[r]; }
        partials[blockIdx.x * 2 + 0] = S;
        partials[blockIdx.x * 2 + 1] = A;
    }
}

// ---------------------------------------------------------------------------
// Kernel 6: deterministic final reduction -> variance[b], loss[b]
// ---------------------------------------------------------------------------
__global__ __launch_bounds__(256) void lv_final(const float* __restrict__ partials,
                                                float* __restrict__ out) {
    __shared__ float sS[256], sA[256];
    int b = blockIdx.x;                       // one block per batch
    float S = 0.0f, A = 0.0f;
    for (int i = threadIdx.x; i < 4608; i += 256) {
        S += partials[(b * 4608 + i) * 2 + 0];
        A += partials[(b * 4608 + i) * 2 + 1];
    }
    sS[threadIdx.x] = S; sA[threadIdx.x] = A;
    __syncthreads();
    for (int st = 128; st; st >>= 1) {
        if (threadIdx.x < st) { sS[threadIdx.x] += sS[threadIdx.x + st];
                                sA[threadIdx.x] += sA[threadIdx.x + st]; }
        __syncthreads();
    }
    if (threadIdx.x == 0) {
        const float invN  = 1.0f / (float)(HWPIX * CC);
        const float sigma = 0.001f * 20.0f / 16.0f;   // 0.00125
        out[b]     = sigma * sS[0] * invN;            // variance
        out[2 + b] = sA[0] * invN;                    // reconstruct_loss
    }
}

// ---------------------------------------------------------------------------
// Launch. Output order: variance(2) | loss(2) | recon(B*HW*C) |
//                       grid_fea(B*K*C) | cond(B*HW)
// Workspace (~672 KB): coef | cond_i | sums | cnt | partials
// ---------------------------------------------------------------------------
extern "C" void kernel_launch(void* const* d_in, const int* in_sizes, int n_in,
                              void* d_out, int out_size, void* d_ws, size_t ws_size,
                              hipStream_t stream) {
    const float* img_fea  = (const float*)d_in[0];
    const float* grid_pos = (const float*)d_in[1];
    const int*   tri      = (const int*)d_in[2];
    float* out = (float*)d_out;

    char* ws = (char*)d_ws;
    float* coef     = (float*)(ws + 0);                         // 36864 B
    int*   cond_i   = (int*)  (ws + 36864);                     // 294912 B
    float* sums     = (float*)(ws + 331776);                    // 262144 B
    int*   cnt      = (int*)  (ws + 593920);                    // 4096 B
    float* partials = (float*)(ws + 598016);                    // 73728 B

    float* out_recon   = out + 4;
    float* out_gridfea = out + 4 + BB * HWPIX * CC;             // 4718596
    float* out_cond    = out_gridfea + BB * KTRI * CC;          // 4784132

    lv_zero   <<<(BB*KTRI*CC + 255) / 256, 256, 0, stream>>>(sums, cnt);
    lv_coef   <<<(BB*KTRI   + 255) / 256, 256, 0, stream>>>(grid_pos, tri, coef);
    lv_cond   <<<BB * 288,              256, 0, stream>>>(coef, cond_i, out_cond);
    lv_accum  <<<(BB*HWPIX*CC) / 256,   256, 0, stream>>>(img_fea, cond_i, sums, cnt);
    lv_gridfea<<<(BB*KTRI*CC)  / 256,   256, 0, stream>>>(sums, cnt, out_gridfea);
    lv_recon  <<<(BB*HWPIX) / 8,        256, 0, stream>>>(img_fea, cond_i, out_gridfea,
                                                          out_recon, partials);
    lv_final  <<<BB,                    256, 0, stream>>>(partials, out);
}